// HunYuanTopKGate_1047972020951
// MI455X (gfx1250) — compile-verified
//
#include <hip/hip_runtime.h>
#include <hip/hip_bf16.h>

typedef float v2f __attribute__((ext_vector_type(2)));
typedef float v4f __attribute__((ext_vector_type(4)));
typedef float v8f __attribute__((ext_vector_type(8)));

#define H_DIM 4096
#define E_NUM 16

// ---------------------------------------------------------------------------
// Kernel 1: logits = x @ wg^T via V_WMMA_F32_16X16X4_F32, then per-token top-2
// Block = 256 threads = 8 waves. Waves 0..7 -> (tileLocal = w&3, ksplit = w>>2).
// Each wave computes a 16-token x 16-expert tile over half of K.
// A layout (16x4 f32): lane<16 holds A[M=lane][K=kb..kb+1] (v0,v1),
//                      lane>=16 holds A[M=lane-16][K=kb+2..kb+3]  -> contiguous float2.
// B layout (4x16 f32): lane<16 holds B[kb][N=lane],B[kb+1][N=lane],
//                      lane>=16 holds B[kb+2][N],B[kb+3][N]; B[k][e]=wg[e][k]
//                      -> contiguous float2 from wg row (lane&15).
// C layout: VGPR j, lane l: C[M = j + 8*(l>>4)][N = l&15].
// ---------------------------------------------------------------------------
__global__ void __launch_bounds__(256)
gate_wmma_kernel(const float* __restrict__ x, const float* __restrict__ wg,
                 int* __restrict__ e0a, int* __restrict__ e1a,
                 float* __restrict__ w0a, float* __restrict__ w1a)
{
    __shared__ float slog[2][64][E_NUM];   // [ksplit][token_local][expert]

    const int wave = threadIdx.x >> 5;
    const int lane = threadIdx.x & 31;
    const int tileLocal = wave & 3;        // 0..3 (16 tokens each)
    const int ksplit    = wave >> 2;       // 0..1
    const int halfK = H_DIM / 2;

    const int rowInTile = lane & 15;
    const int kPair     = (lane >> 4) << 1;        // 0 or 2
    const int token0    = blockIdx.x * 64 + tileLocal * 16;

    const float* __restrict__ xptr =
        x + (long)(token0 + rowInTile) * H_DIM + ksplit * halfK + kPair;
    const float* __restrict__ wptr =
        wg + (long)rowInTile * H_DIM + ksplit * halfK + kPair;

    v8f acc = {};
#pragma unroll 8
    for (int kb = 0; kb < halfK; kb += 4) {
        v2f a = *(const v2f*)(xptr + kb);
        v2f b = *(const v2f*)(wptr + kb);
        acc = __builtin_amdgcn_wmma_f32_16x16x4_f32(
            /*neg_a=*/false, a, /*neg_b=*/false, b,
            /*c_mod=*/(short)0, acc, /*reuse_a=*/false, /*reuse_b=*/false);
    }

    const int mHi = (lane >> 4) << 3;      // 0 or 8
#pragma unroll
    for (int j = 0; j < 8; ++j)
        slog[ksplit][tileLocal * 16 + j + mHi][lane & 15] = acc[j];

    __syncthreads();

    // per-token top-2 (64 tokens per block, one thread each)
    if (threadIdx.x < 64) {
        const int tl = threadIdx.x;
        float L[E_NUM];
#pragma unroll
        for (int e = 0; e < E_NUM; ++e)
            L[e] = slog[0][tl][e] + slog[1][tl][e];

        int e0 = 0; float l0 = L[0];
#pragma unroll
        for (int e = 1; e < E_NUM; ++e)
            if (L[e] > l0) { l0 = L[e]; e0 = e; }
        int e1 = (e0 == 0) ? 1 : 0; float l1 = L[e1];
#pragma unroll
        for (int e = 0; e < E_NUM; ++e)
            if (e != e0 && L[e] > l1) { l1 = L[e]; e1 = e; }

        // softmax denominator cancels: w0 = g0/(g0+g1) = 1/(1+exp(l1-l0))
        float w0 = 1.0f / (1.0f + expf(l1 - l0));

        const int t = blockIdx.x * 64 + tl;
        e0a[t] = e0; e1a[t] = e1;
        w0a[t] = w0; w1a[t] = 1.0f - w0;
    }
}

// ---------------------------------------------------------------------------
// Kernel 2: token priorities (per-expert prefix counts in flat order
// k*T + t). Single workgroup: 128 threads x TPT tokens each, chunked
// histogram + cross-thread exclusive scan; k=0 totals seed k=1 base.
// ---------------------------------------------------------------------------
__global__ void __launch_bounds__(128)
priority_kernel(const int* __restrict__ e0a, const int* __restrict__ e1a,
                int* __restrict__ p0a, int* __restrict__ p1a, int T)
{
    __shared__ int hist[128][E_NUM];
    __shared__ int base[E_NUM];

    const int tid = threadIdx.x;
    const int TPT = T / 128;
    const int start = tid * TPT;

    // ---- phase 0 (k = 0) ----
    int cnt[E_NUM];
#pragma unroll
    for (int e = 0; e < E_NUM; ++e) cnt[e] = 0;
    for (int i = 0; i < TPT; ++i) cnt[e0a[start + i]]++;
#pragma unroll
    for (int e = 0; e < E_NUM; ++e) hist[tid][e] = cnt[e];
    __syncthreads();

    int off[E_NUM];
#pragma unroll
    for (int e = 0; e < E_NUM; ++e) off[e] = 0;
    for (int tp = 0; tp < tid; ++tp)
#pragma unroll
        for (int e = 0; e < E_NUM; ++e) off[e] += hist[tp][e];

    if (tid == 127)
#pragma unroll
        for (int e = 0; e < E_NUM; ++e) base[e] = off[e] + cnt[e];

    for (int i = 0; i < TPT; ++i) {
        int e = e0a[start + i];
        p0a[start + i] = off[e]++;
    }
    __syncthreads();   // everyone done reading phase-0 hist; base[] visible

    // ---- phase 1 (k = 1), base = total k=0 counts ----
#pragma unroll
    for (int e = 0; e < E_NUM; ++e) cnt[e] = 0;
    for (int i = 0; i < TPT; ++i) cnt[e1a[start + i]]++;
#pragma unroll
    for (int e = 0; e < E_NUM; ++e) hist[tid][e] = cnt[e];
    __syncthreads();

#pragma unroll
    for (int e = 0; e < E_NUM; ++e) off[e] = base[e];
    for (int tp = 0; tp < tid; ++tp)
#pragma unroll
        for (int e = 0; e < E_NUM; ++e) off[e] += hist[tp][e];

    for (int i = 0; i < TPT; ++i) {
        int e = e1a[start + i];
        p1a[start + i] = off[e]++;
    }
}

// ---------------------------------------------------------------------------
// Kernel 3: zero-fill the 268 MB output with nontemporal 128-bit stores.
// ---------------------------------------------------------------------------
__global__ void __launch_bounds__(256)
zerofill_kernel(v4f* __restrict__ out, long n4)
{
    long i = (long)blockIdx.x * blockDim.x + threadIdx.x;
    const long stride = (long)gridDim.x * blockDim.x;
    v4f z = {};
    for (; i < n4; i += stride)
        __builtin_nontemporal_store(z, out + i);
}

// ---------------------------------------------------------------------------
// Kernel 4: scatter the 2*T nonzeros into combine_weights / dispatch_mask.
// ---------------------------------------------------------------------------
__global__ void __launch_bounds__(256)
scatter_kernel(const int* __restrict__ e0a, const int* __restrict__ e1a,
               const int* __restrict__ p0a, const int* __restrict__ p1a,
               const float* __restrict__ w0a, const float* __restrict__ w1a,
               float* __restrict__ out, int T, int cap)
{
    int idx = blockIdx.x * blockDim.x + threadIdx.x;
    if (idx >= 2 * T) return;
    const int k = idx / T;
    const int t = idx - k * T;
    const int e = k ? e1a[t] : e0a[t];
    const int p = k ? p1a[t] : p0a[t];
    const float w = k ? w1a[t] : w0a[t];
    if (p < cap) {
        long o = ((long)t * E_NUM + e) * cap + p;
        out[o] = w;                               // combine_weights
        out[o + (long)T * E_NUM * cap] = 1.0f;    // dispatch_mask (as float)
    }
}

extern "C" void kernel_launch(void* const* d_in, const int* in_sizes, int n_in,
                              void* d_out, int out_size, void* d_ws, size_t ws_size,
                              hipStream_t stream)
{
    const float* x  = (const float*)d_in[0];   // [2,2048,4096] -> [T,H]
    const float* wg = (const float*)d_in[1];   // [16,4096]

    const int T = in_sizes[0] / H_DIM;         // 4096
    int cap = (2 * T) / E_NUM;                 // 512
    if (cap < 2) cap = 2;

    // workspace layout
    int*   e0 = (int*)d_ws;
    int*   e1 = e0 + T;
    float* w0 = (float*)(e1 + T);
    float* w1 = w0 + T;
    int*   p0 = (int*)(w1 + T);
    int*   p1 = p0 + T;

    // 1) GEMM gate + top-2 (WMMA): 64 tokens / block
    gate_wmma_kernel<<<T / 64, 256, 0, stream>>>(x, wg, e0, e1, w0, w1);

    // 2) priorities (single workgroup scan)
    priority_kernel<<<1, 128, 0, stream>>>(e0, e1, p0, p1, T);

    // 3) zero-fill outputs (combine + dispatch, concatenated)
    long n4 = (long)out_size / 4;
    zerofill_kernel<<<2048, 256, 0, stream>>>((v4f*)d_out, n4);

    // 4) scatter nonzeros
    scatter_kernel<<<(2 * T + 255) / 256, 256, 0, stream>>>(
        e0, e1, p0, p1, w0, w1, (float*)d_out, T, cap);
}